// WeightedSegmentationLoss_53489522704550
// MI455X (gfx1250) — compile-verified
//
#include <hip/hip_runtime.h>
#include <hip/hip_bf16.h>
#include <stdint.h>

#define B_ 16
#define C_ 6
#define H_ 512
#define W_ 512
#define BIGV 1e10f

typedef float v2f __attribute__((ext_vector_type(2)));
typedef float v8f __attribute__((ext_vector_type(8)));
typedef unsigned int v4u __attribute__((ext_vector_type(4)));
typedef int v4i __attribute__((ext_vector_type(4)));
typedef int v8i __attribute__((ext_vector_type(8)));

// ---------------------------------------------------------------------------
// Pass A: exact 1D chamfer distance along H per (b,w) column, squared result.
// One thread per column; accesses coalesced across w.
// ---------------------------------------------------------------------------
__global__ __launch_bounds__(256) void edt_cols_kernel(
    const int* __restrict__ target, float* __restrict__ d1sq) {
  const int tid = blockIdx.x * 256 + threadIdx.x;  // 0 .. B*W-1
  const int b = tid >> 9;
  const int w = tid & (W_ - 1);
  const int* tcol = target + (size_t)b * H_ * W_ + w;
  float* dcol = d1sq + (size_t)b * H_ * W_ + w;

  float c = BIGV;
#pragma unroll 4
  for (int y = 0; y < H_; ++y) {
    float f = (tcol[(size_t)y * W_] < (C_ - 1)) ? 0.0f : BIGV;
    c = fminf(f, c + 1.0f);
    dcol[(size_t)y * W_] = c;  // forward chamfer stored temporarily
  }
  float c2 = BIGV;
#pragma unroll 4
  for (int y = H_ - 1; y >= 0; --y) {
    float f = (tcol[(size_t)y * W_] < (C_ - 1)) ? 0.0f : BIGV;
    c2 = fminf(f, c2 + 1.0f);
    float d = fminf(dcol[(size_t)y * W_], c2);
    dcol[(size_t)y * W_] = d * d;  // d1^2
  }
}

// ---------------------------------------------------------------------------
// Pass B: one workgroup per (b,h) row.
//   * TDM DMA of the 6x512 strided class tile pred[b, :, h, :] into LDS,
//     overlapped with the exact O(W^2) squared-EDT min-scan along W.
//   * Then weight = exp(-dist2/200), 6-class log-softmax CE, block reduction.
// ---------------------------------------------------------------------------
struct SMem {
  float tile[C_ * W_];  // 12 KB : TDM destination (pred tile, class-major)
  float row[W_];        //  2 KB : d1^2 row
  float red[16];
};

__global__ __launch_bounds__(512) void row_fused_kernel(
    const float* __restrict__ pred, const int* __restrict__ target,
    const float* __restrict__ d1sq, float* __restrict__ partials) {
  __shared__ SMem sm;
  const int x = threadIdx.x;            // 0..511
  const int b = blockIdx.x >> 9;
  const int h = blockIdx.x & (H_ - 1);

  // ---- issue Tensor Data Mover load (wave 0 only) -------------------------
  if (x < 32) {
    const float* tile0 = pred + ((size_t)b * C_ * H_ + h) * W_;  // pred[b,0,h,0]
    unsigned long long ga = (unsigned long long)(uintptr_t)tile0;
    unsigned ldsa = (unsigned)(uintptr_t)(void*)sm.tile;
    v4u g0;
    g0[0] = 1u;                                                // count=1
    g0[1] = ldsa;                                              // lds_addr
    g0[2] = (unsigned)(ga & 0xFFFFFFFFull);                    // global_addr lo
    g0[3] = (unsigned)((ga >> 32) & 0x01FFFFFFull) | (2u << 30);  // hi | type=2
    v8i g1;
    g1[0] = (int)(2u << 16);                // data_size = 4 bytes
    g1[1] = (int)((unsigned)W_ << 16);      // tensor_dim0 = 512 (bits 79:48 lo)
    g1[2] = (int)((unsigned)C_ << 16);      // tensor_dim1 = 6
    g1[3] = (int)((unsigned)W_ << 16);      // tile_dim0 = 512
    g1[4] = C_;                             // tile_dim1 = 6, tile_dim2 = 0
    g1[5] = H_ * W_;                        // tensor_dim0_stride = 262144 elems
    g1[6] = 0;                              // stride hi / dim1_stride lo
    g1[7] = 0;
    v4i gz = {0, 0, 0, 0};
    v8i gz8 = {0, 0, 0, 0, 0, 0, 0, 0};
    __builtin_amdgcn_tensor_load_to_lds(g0, g1, gz, gz, gz8, 0);
  }

  // ---- stage d1^2 row into LDS (overlaps with TDM) ------------------------
  sm.row[x] = d1sq[((size_t)b * H_ + h) * W_ + x];
  __syncthreads();

  // ---- exact squared EDT along W: min_xp ( row[xp] + (x-xp)^2 ) -----------
  float d2 = 1e30f;
  float diff = (float)x;  // x - xp, xp = 0
#pragma unroll 4
  for (int xp = 0; xp < W_; xp += 4) {
    float4 q = *reinterpret_cast<const float4*>(&sm.row[xp]);  // broadcast b128
    d2 = fminf(d2, fmaf(diff, diff, q.x)); diff -= 1.0f;
    d2 = fminf(d2, fmaf(diff, diff, q.y)); diff -= 1.0f;
    d2 = fminf(d2, fmaf(diff, diff, q.z)); diff -= 1.0f;
    d2 = fminf(d2, fmaf(diff, diff, q.w)); diff -= 1.0f;
  }
  float wgt = __expf(d2 * (-1.0f / 200.0f));  // exp(-dist2 / (2*sigma^2))

  // ---- wait for TDM tile, then cross-entropy ------------------------------
  if (x < 32) __builtin_amdgcn_s_wait_tensorcnt(0);
  __syncthreads();

  float l0 = sm.tile[0 * W_ + x], l1 = sm.tile[1 * W_ + x];
  float l2 = sm.tile[2 * W_ + x], l3 = sm.tile[3 * W_ + x];
  float l4 = sm.tile[4 * W_ + x], l5 = sm.tile[5 * W_ + x];
  float m = fmaxf(fmaxf(fmaxf(l0, l1), fmaxf(l2, l3)), fmaxf(l4, l5));
  float s = __expf(l0 - m) + __expf(l1 - m) + __expf(l2 - m) +
            __expf(l3 - m) + __expf(l4 - m) + __expf(l5 - m);
  float lse = m + __logf(s);
  int t = target[((size_t)b * H_ + h) * W_ + x];
  float lt = l5;
  lt = (t == 0) ? l0 : lt;
  lt = (t == 1) ? l1 : lt;
  lt = (t == 2) ? l2 : lt;
  lt = (t == 3) ? l3 : lt;
  lt = (t == 4) ? l4 : lt;
  float v = wgt * (lse - lt);

  // ---- deterministic block reduction --------------------------------------
#pragma unroll
  for (int off = 16; off > 0; off >>= 1) v += __shfl_down(v, off);
  if ((x & 31) == 0) sm.red[x >> 5] = v;
  __syncthreads();
  if (x < 32) {
    float r = (x < 16) ? sm.red[x] : 0.0f;
#pragma unroll
    for (int off = 8; off > 0; off >>= 1) r += __shfl_down(r, off);
    if (x == 0) partials[blockIdx.x] = r;
  }
}

// ---------------------------------------------------------------------------
// Pass C: 8192 partials -> scalar mean. Single wave; fixed-order lane sums,
// then a WMMA ones-matrix reduction (D[m,n] = s[m] + s[m+16]) in full fp32.
// ---------------------------------------------------------------------------
__global__ __launch_bounds__(32) void final_reduce_kernel(
    const float* __restrict__ partials, float* __restrict__ out) {
  const int lane = threadIdx.x;  // 0..31
  float s = 0.0f;
  for (int i = lane; i < B_ * H_; i += 32) s += partials[i];

  // A (16x4 f32): VGPR0 = s -> A[m,0]=s[m], A[m,2]=s[m+16]; VGPR1 = 0.
  // B (4x16 f32) = all ones  =>  D[m,n] = s[m] + s[m+16] for every n.
  v2f a;  a[0] = s;    a[1] = 0.0f;
  v2f bo; bo[0] = 1.0f; bo[1] = 1.0f;
  v8f c = {};
  v8f d = __builtin_amdgcn_wmma_f32_16x16x4_f32(
      /*neg_a=*/false, a, /*neg_b=*/false, bo,
      /*c_mod=*/(short)0, c, /*reuse_a=*/false, /*reuse_b=*/false);
  // lane 0 holds D[0..7, 0]; lane 16 holds D[8..15, 0].
  float t = d[0] + d[1] + d[2] + d[3] + d[4] + d[5] + d[6] + d[7];
  t += __shfl_down(t, 16);
  if (lane == 0) out[0] = t * (1.0f / (float)((size_t)B_ * H_ * W_));
}

// ---------------------------------------------------------------------------
extern "C" void kernel_launch(void* const* d_in, const int* in_sizes, int n_in,
                              void* d_out, int out_size, void* d_ws,
                              size_t ws_size, hipStream_t stream) {
  const float* pred = (const float*)d_in[0];    // [16, 6, 512, 512] fp32
  const int* target = (const int*)d_in[1];      // [16, 512, 512] int
  float* out = (float*)d_out;                   // scalar fp32

  float* d1sq = (float*)d_ws;                              // 16 MB
  float* partials = d1sq + (size_t)B_ * H_ * W_;           // + 32 KB

  edt_cols_kernel<<<(B_ * W_) / 256, 256, 0, stream>>>(target, d1sq);
  row_fused_kernel<<<B_ * H_, 512, 0, stream>>>(pred, target, d1sq, partials);
  final_reduce_kernel<<<1, 32, 0, stream>>>(partials, out);
}